// LinearClassifier_32323923870270
// MI455X (gfx1250) — compile-verified
//
#include <hip/hip_runtime.h>

// CDNA5 (gfx1250) wave32 WMMA types
typedef float v2f __attribute__((ext_vector_type(2)));
typedef float v8f __attribute__((ext_vector_type(8)));

#define BATCH   4096
#define N_DIM   2048
#define N_CLASS 1000

// GEMM tiling: 256 threads = 8 waves arranged 2(M) x 4(N); each wave owns a
// 32x32 macro-tile (2x2 grid of 16x16 WMMA tiles) -> block tile 64 x 128.
#define BM 64
#define BN 128
#define BK 64
// LDS row stride: 68 floats -> fragment-read bank = (4*row + k) % 64,
// conflict-free across all 32 lanes; 68*4 = 272 B keeps rows 16B-aligned.
#define LDST 68

// ---------------------------------------------------------------------------
// Kernel 1: x_mean[b,d] = mean_t x[b,d,t].  Pure bandwidth: each lane reads a
// contiguous 64B (4 x float4) run; this is the 512 MiB / ~23 us HBM floor.
// ---------------------------------------------------------------------------
__global__ __launch_bounds__(256)
void mean16_kernel(const float* __restrict__ x, float* __restrict__ xm) {
    size_t i = (size_t)blockIdx.x * blockDim.x + threadIdx.x; // [0, 4096*2048)
    const float4* p = (const float4*)(x + i * 16);
    float4 a = p[0], b = p[1], c = p[2], d = p[3];
    float s = ((a.x + a.y) + (a.z + a.w)) + ((b.x + b.y) + (b.z + b.w)) +
              ((c.x + c.y) + (c.z + c.w)) + ((d.x + d.y) + (d.z + d.w));
    xm[i] = s * (1.0f / 16.0f);
}

// ---------------------------------------------------------------------------
// Kernel 2: out[m,n] = sum_k xm[m,k] * W[n,k] + bias[n]
// f32 WMMA (v_wmma_f32_16x16x4_f32) keeps exact fp32 semantics; W lives in L2.
// 2x2 register blocking: 4 WMMAs per 2 A-frags + 2 B-frags.
// ---------------------------------------------------------------------------
__global__ __launch_bounds__(256)
void gemm_wmma_kernel(const float* __restrict__ A,    // xm   [4096, 2048]
                      const float* __restrict__ W,    // W    [1000, 2048]
                      const float* __restrict__ bias, // b    [1000]
                      float* __restrict__ out) {      // out  [4096, 1000]
    __shared__ float As[BM * LDST];   // 64 x 68 f32  (~17.4 KB)
    __shared__ float Bs[BN * LDST];   // 128 x 68 f32 (~34.8 KB)

    const int tid  = threadIdx.x;
    const int lane = tid & 31;
    const int wave = tid >> 5;      // 0..7
    const int mw   = wave >> 2;     // 0..1  -> 32-row macro-tile
    const int nw   = wave & 3;      // 0..3  -> 32-col macro-tile

    const int mbase = blockIdx.x * BM;
    const int nbase = blockIdx.y * BN;

    // ISA 32-bit 16x4 A layout: lanes 0-15 -> K={0,1}, lanes 16-31 -> K={2,3};
    // B (4x16) mirrors it with N striped across lanes.
    const int l16   = lane & 15;
    const int khalf = (lane >> 4) << 1;       // 0 or 2
    const int ar0 = (mw * 32 + l16) * LDST;   // A rows for tile m0
    const int ar1 = ar0 + 16 * LDST;          // A rows for tile m1
    const int br0 = (nw * 32 + l16) * LDST;   // B rows for tile n0
    const int br1 = br0 + 16 * LDST;          // B rows for tile n1

    v8f acc00 = {}, acc01 = {}, acc10 = {}, acc11 = {};

    for (int kb = 0; kb < N_DIM; kb += BK) {
        // Stage A tile: 64x64 f32 = 1024 float4, 4 per thread (coalesced rows)
        #pragma unroll
        for (int r = 0; r < 4; ++r) {
            int q   = tid + r * 256;
            int row = q >> 4;
            int c4  = (q & 15) << 2;
            float4 v = *(const float4*)(&A[(size_t)(mbase + row) * N_DIM + kb + c4]);
            *(float4*)(&As[row * LDST + c4]) = v;
        }
        // Stage B tile: 128x64 f32 = 2048 float4, 8 per thread; rows n>=1000 -> 0
        #pragma unroll
        for (int r = 0; r < 8; ++r) {
            int q   = tid + r * 256;
            int row = q >> 4;
            int c4  = (q & 15) << 2;
            int n   = nbase + row;
            float4 v = make_float4(0.f, 0.f, 0.f, 0.f);
            if (n < N_CLASS)
                v = *(const float4*)(&W[(size_t)n * N_DIM + kb + c4]);
            *(float4*)(&Bs[row * LDST + c4]) = v;
        }
        __syncthreads();

        // 16 K-steps of 4; 4 WMMAs per step with 2x2 fragment reuse.
        // EXEC is all-ones here (guarded loads reconverged above).
        #pragma unroll
        for (int kk = 0; kk < BK; kk += 4) {
            v2f a0 = *(const v2f*)(&As[ar0 + kk + khalf]);
            v2f a1 = *(const v2f*)(&As[ar1 + kk + khalf]);
            v2f b0 = *(const v2f*)(&Bs[br0 + kk + khalf]);
            v2f b1 = *(const v2f*)(&Bs[br1 + kk + khalf]);
            acc00 = __builtin_amdgcn_wmma_f32_16x16x4_f32(false, a0, false, b0,
                                                          (short)0, acc00, false, false);
            acc01 = __builtin_amdgcn_wmma_f32_16x16x4_f32(false, a0, false, b1,
                                                          (short)0, acc01, false, false);
            acc10 = __builtin_amdgcn_wmma_f32_16x16x4_f32(false, a1, false, b0,
                                                          (short)0, acc10, false, false);
            acc11 = __builtin_amdgcn_wmma_f32_16x16x4_f32(false, a1, false, b1,
                                                          (short)0, acc11, false, false);
        }
        __syncthreads();
    }

    // Epilogue: C/D layout -> VGPR i holds rows {i, i+8}, N = lane%16.
    const int rb   = mbase + mw * 32 + ((lane >> 4) << 3);  // + i, +16 for m1
    const int n0   = nbase + nw * 32 + l16;
    const int n1   = n0 + 16;
    const float bv0 = (n0 < N_CLASS) ? bias[n0] : 0.0f;
    const float bv1 = (n1 < N_CLASS) ? bias[n1] : 0.0f;
    #pragma unroll
    for (int i = 0; i < 8; ++i) {
        if (n0 < N_CLASS) {
            out[(size_t)(rb + i) * N_CLASS + n0]      = acc00[i] + bv0;
            out[(size_t)(rb + 16 + i) * N_CLASS + n0] = acc10[i] + bv0;
        }
        if (n1 < N_CLASS) {
            out[(size_t)(rb + i) * N_CLASS + n1]      = acc01[i] + bv1;
            out[(size_t)(rb + 16 + i) * N_CLASS + n1] = acc11[i] + bv1;
        }
    }
}

// ---------------------------------------------------------------------------
extern "C" void kernel_launch(void* const* d_in, const int* in_sizes, int n_in,
                              void* d_out, int out_size, void* d_ws, size_t ws_size,
                              hipStream_t stream) {
    const float* x  = (const float*)d_in[0];   // [4096, 2048, 16]
    const float* W  = (const float*)d_in[1];   // [1000, 2048]
    const float* b  = (const float*)d_in[2];   // [1000]
    float* out = (float*)d_out;                // [4096, 1000]
    float* xm  = (float*)d_ws;                 // 4096*2048 f32 = 32 MiB scratch

    mean16_kernel<<<(BATCH * N_DIM) / 256, 256, 0, stream>>>(x, xm);

    dim3 grid(BATCH / BM, (N_CLASS + BN - 1) / BN);  // 64 x 8
    gemm_wmma_kernel<<<grid, 256, 0, stream>>>(xm, W, b, out);
}